// HT_2413771621187
// MI455X (gfx1250) — compile-verified
//
#include <hip/hip_runtime.h>

// ---------------------------------------------------------------------------
// HT layer:  y_b = sum_p WLt_p (64x64, ac x ij) * X_b (64x64, ij x kl) * WR_p
// (64x64, kl x de), then + bias.  f32 WMMA 16x16x4 on gfx1250 (wave32).
// ---------------------------------------------------------------------------

typedef float v2f __attribute__((ext_vector_type(2)));
typedef float v8f __attribute__((ext_vector_type(8)));

#define FEAT 4096
#define LDSP 68              // padded LDS row stride (floats), multiple of 4
#define PSTRIDE (64 * LDSP)  // one 64x64 padded matrix

__device__ __forceinline__ v8f wmma_f32(v2f a, v2f b, v8f c) {
    // 8 args: (neg_a, A, neg_b, B, c_mod, C, reuse_a, reuse_b)
    return __builtin_amdgcn_wmma_f32_16x16x4_f32(false, a, false, b,
                                                 (short)0, c, false, false);
}

// ---------------------------------------------------------------------------
// Prep: WLt[p][ac][ij] = sum_{r,s} F0[i,a,r] F1[j,c,s] CL[r,s,p]
//       WR [p][kl][de] = sum_{r,s} F2[k,d,r] F3[l,e,s] CR[r,s,p]
// 65536 threads total (first 32768 -> WL, rest -> WR).
// ---------------------------------------------------------------------------
__global__ __launch_bounds__(256) void ht_prep(
    const float* __restrict__ F0, const float* __restrict__ F1,
    const float* __restrict__ F2, const float* __restrict__ F3,
    const float* __restrict__ CL, const float* __restrict__ CR,
    float* __restrict__ WLo, float* __restrict__ WRo) {
    int t = blockIdx.x * 256 + threadIdx.x;
    bool isR = (t >= 32768);
    int u   = isR ? (t - 32768) : t;
    int p   = u >> 12;
    int rem = u & 4095;
    int row = rem >> 6;   // ac (WL) or kl (WR)
    int col = rem & 63;   // ij (WL) or de (WR)
    float acc = 0.0f;
    if (!isR) {
        int a = row >> 3, c = row & 7, i = col >> 3, j = col & 7;
        for (int r = 0; r < 8; ++r) {
            float f0 = F0[(i * 8 + a) * 8 + r];
            for (int s = 0; s < 8; ++s)
                acc += f0 * F1[(j * 8 + c) * 8 + s] * CL[(r * 8 + s) * 8 + p];
        }
        WLo[u] = acc;
    } else {
        int k = row >> 3, l = row & 7, d = col >> 3, e = col & 7;
        for (int r = 0; r < 8; ++r) {
            float f2 = F2[(k * 8 + d) * 8 + r];
            for (int s = 0; s < 8; ++s)
                acc += f2 * F3[(l * 8 + e) * 8 + s] * CR[(r * 8 + s) * 8 + p];
        }
        WRo[u] = acc;
    }
}

// ---------------------------------------------------------------------------
// Main kernel: 256 threads = 8 wave32.  Dynamic LDS layout (floats):
//   wls[8][64][LDSP] | wrs[8][64][LDSP] | xs[64][LDSP] | ts[64][LDSP]
// Per wave: 2 of the 16 16x16 output tiles (both stages).
// ---------------------------------------------------------------------------
__global__ __launch_bounds__(256) void ht_main(
    const float* __restrict__ x, const float* __restrict__ WL,
    const float* __restrict__ WR, const float* __restrict__ bias,
    float* __restrict__ out, int B) {
    extern __shared__ float smem[];
    float* wls = smem;                  // 8*64*68
    float* wrs = wls + 8 * PSTRIDE;     // 8*64*68
    float* xs  = wrs + 8 * PSTRIDE;     // 64*68  (X_b, [ij][kl])
    float* ts  = xs + PSTRIDE;          // 64*68  (T_p, [ac][kl])

    const int tid  = threadIdx.x;
    const int lane = tid & 31;
    const int wv   = tid >> 5;
    const int lm   = lane & 15;  // tile row (A) / tile col (B/C/D)
    const int lh   = lane >> 4;  // lane half selects K pair / M half

    // Load both weight stacks into padded LDS (constant across all b).
    for (int t = tid; t < 8 * 64 * 64; t += 256) {
        int p = t >> 12, rem = t & 4095, r = rem >> 6, c = rem & 63;
        int dst = p * PSTRIDE + r * LDSP + c;
        wls[dst] = WL[t];
        wrs[dst] = WR[t];
    }
    __syncthreads();

    // Tile assignment: tiles 2*wv, 2*wv+1 of 16; tile t -> rows (t>>2)*16, cols (t&3)*16
    const int t0 = 2 * wv, t1 = 2 * wv + 1;
    const int ar0 = (t0 >> 2) * 16, cc0 = (t0 & 3) * 16;
    const int ar1 = (t1 >> 2) * 16, cc1 = (t1 & 3) * 16;

    for (int b = blockIdx.x; b < B; b += gridDim.x) {
        // Stage X_b into LDS, coalesced float4 (identity layout + pad).
        const float4* xsrc = (const float4*)(x + (size_t)b * FEAT);
        for (int v = tid; v < 1024; v += 256) {
            float4 q = xsrc[v];
            int ij = v >> 4, kl = (v & 15) << 2;
            *(float4*)&xs[ij * LDSP + kl] = q;
        }
        // Prefetch next batch row while we compute this one.
        if (b + (int)gridDim.x < B) {
            const float* nx = x + (size_t)(b + gridDim.x) * FEAT;
            __builtin_prefetch(nx + tid * 16, 0, 1);
        }
        __syncthreads();

        v8f y0 = {0.f, 0.f, 0.f, 0.f, 0.f, 0.f, 0.f, 0.f};
        v8f y1 = {0.f, 0.f, 0.f, 0.f, 0.f, 0.f, 0.f, 0.f};

        for (int p = 0; p < 8; ++p) {
            const float* wlp = wls + p * PSTRIDE;
            const float* wrp = wrs + p * PSTRIDE;

            // ---- Stage 1: T_p = WLt_p * X_b -----------------------------
            v8f ta = {0.f, 0.f, 0.f, 0.f, 0.f, 0.f, 0.f, 0.f};
            v8f tb = {0.f, 0.f, 0.f, 0.f, 0.f, 0.f, 0.f, 0.f};
#pragma unroll
            for (int kk = 0; kk < 16; ++kk) {
                int k0 = kk * 4 + 2 * lh;  // this lane-half's K pair
                v2f a0, b0, a1, b1;
                a0.x = wlp[(ar0 + lm) * LDSP + k0];
                a0.y = wlp[(ar0 + lm) * LDSP + k0 + 1];
                b0.x = xs[k0 * LDSP + cc0 + lm];
                b0.y = xs[(k0 + 1) * LDSP + cc0 + lm];
                ta = wmma_f32(a0, b0, ta);
                a1.x = wlp[(ar1 + lm) * LDSP + k0];
                a1.y = wlp[(ar1 + lm) * LDSP + k0 + 1];
                b1.x = xs[k0 * LDSP + cc1 + lm];
                b1.y = xs[(k0 + 1) * LDSP + cc1 + lm];
                tb = wmma_f32(a1, b1, tb);
            }
            // Store T tiles: VGPR r holds M = r + 8*lh, N = lm.
#pragma unroll
            for (int r = 0; r < 8; ++r) {
                ts[(ar0 + r + 8 * lh) * LDSP + cc0 + lm] = ta[r];
                ts[(ar1 + r + 8 * lh) * LDSP + cc1 + lm] = tb[r];
            }
            __syncthreads();

            // ---- Stage 2: y += T_p * WR_p -------------------------------
#pragma unroll
            for (int kk = 0; kk < 16; ++kk) {
                int k0 = kk * 4 + 2 * lh;
                v2f a0, b0, a1, b1;
                a0.x = ts[(ar0 + lm) * LDSP + k0];
                a0.y = ts[(ar0 + lm) * LDSP + k0 + 1];
                b0.x = wrp[k0 * LDSP + cc0 + lm];
                b0.y = wrp[(k0 + 1) * LDSP + cc0 + lm];
                y0 = wmma_f32(a0, b0, y0);
                a1.x = ts[(ar1 + lm) * LDSP + k0];
                a1.y = ts[(ar1 + lm) * LDSP + k0 + 1];
                b1.x = wrp[k0 * LDSP + cc1 + lm];
                b1.y = wrp[(k0 + 1) * LDSP + cc1 + lm];
                y1 = wmma_f32(a1, b1, y1);
            }
            __syncthreads();  // ts is rewritten next p-iteration
        }

        // Write out with fused bias (rows stride 64 floats, 16-lane runs).
        float* outb = out + (size_t)b * FEAT;
#pragma unroll
        for (int r = 0; r < 8; ++r) {
            int row0 = ar0 + r + 8 * lh, idx0 = row0 * 64 + cc0 + lm;
            outb[idx0] = y0[r] + bias[idx0];
            int row1 = ar1 + r + 8 * lh, idx1 = row1 * 64 + cc1 + lm;
            outb[idx1] = y1[r] + bias[idx1];
        }
        // no barrier needed: next iteration's xs store is ordered behind the
        // final p-loop barrier, after which no wave reads xs or ts.
    }
}

// ---------------------------------------------------------------------------
extern "C" void kernel_launch(void* const* d_in, const int* in_sizes, int n_in,
                              void* d_out, int out_size, void* d_ws, size_t ws_size,
                              hipStream_t stream) {
    const float* x    = (const float*)d_in[0];
    const float* F0   = (const float*)d_in[1];
    const float* F1   = (const float*)d_in[2];
    const float* F2   = (const float*)d_in[3];
    const float* F3   = (const float*)d_in[4];
    const float* CL   = (const float*)d_in[5];
    const float* CR   = (const float*)d_in[6];
    const float* bias = (const float*)d_in[7];
    float* out = (float*)d_out;

    int B = in_sizes[0] / FEAT;

    float* wl = (float*)d_ws;          // 8*64*64 floats
    float* wr = wl + 8 * 64 * 64;      // 8*64*64 floats

    ht_prep<<<256, 256, 0, stream>>>(F0, F1, F2, F3, CL, CR, wl, wr);

    size_t shbytes = (size_t)(2 * 8 * PSTRIDE + 2 * PSTRIDE) * sizeof(float); // ~306 KB
    (void)hipFuncSetAttribute((const void*)ht_main,
                              hipFuncAttributeMaxDynamicSharedMemorySize,
                              (int)shbytes);
    ht_main<<<512, 256, shbytes, stream>>>(x, wl, wr, bias, out, B);
}